// HeteroSageEncoder_8083128451627
// MI455X (gfx1250) — compile-verified
//
#include <hip/hip_runtime.h>

typedef __attribute__((ext_vector_type(2))) float v2f;
typedef __attribute__((ext_vector_type(8))) float v8f;

#define NN   100000
#define DIN_ 128
#define DOUT_ 256
#define RR   4
#define EE   600000

#define KCH 320   // K-chunk staged in LDS (divides 640 and 1280)
#define LDA 322   // padded LDS row stride (floats) to reduce bank conflicts

// ---------------- utility kernels ----------------

__global__ void zero_kernel(float4* p, long n4) {
  long i = (long)blockIdx.x * blockDim.x + threadIdx.x;
  long stride = (long)gridDim.x * blockDim.x;
  float4 z; z.x = 0.f; z.y = 0.f; z.z = 0.f; z.w = 0.f;
  for (; i < n4; i += stride) p[i] = z;
}

__global__ void bias_sum_kernel(const float* __restrict__ b, float* __restrict__ out) {
  int o = blockIdx.x * blockDim.x + threadIdx.x;
  if (o < DOUT_) {
    float s = 0.f;
    for (int r = 0; r < RR; ++r) s += b[r * DOUT_ + o];
    out[o] = s;
  }
}

// Build B in WMMA-fragment order. Logical combined weight:
//   Wcat[o][k] = Wn[r=k/D][o][k%D]           for k <  R*D
//             = sum_r Wr[r][o][k-R*D]        for k >= R*D   (mean over r folded as sum; 1/R applied later)
// Fragment layout (V_WMMA_F32_16X16X4_F32 B operand, 4x16 per step):
//   Bswz[((ntile*ksteps + kk)*32 + lane)*2 + c] = Wcat[ntile*16 + lane%16][kk*4 + (lane/16)*2 + c]
__global__ void build_bswz_kernel(const float* __restrict__ Wn, const float* __restrict__ Wr,
                                  int D, float* __restrict__ Bswz) {
  int K = RR * D + D;
  int ksteps = K >> 2;
  int total = 16 * ksteps * 64;
  int gid = blockIdx.x * blockDim.x + threadIdx.x;
  if (gid >= total) return;
  int c     = gid & 1;
  int lane  = (gid >> 1) & 31;
  int kk    = (gid >> 6) % ksteps;
  int ntile = (gid >> 6) / ksteps;
  int k = kk * 4 + ((lane >> 4) << 1) + c;
  int o = ntile * 16 + (lane & 15);
  float v;
  if (k < RR * D) {
    int r = k / D, d = k - r * D;
    v = Wn[((size_t)r * DOUT_ + o) * D + d];
  } else {
    int d = k - RR * D;
    v = 0.f;
    for (int r = 0; r < RR; ++r) v += Wr[((size_t)r * DOUT_ + o) * D + d];
  }
  Bswz[gid] = v;
}

// One wave per edge: gather feat[src] (D floats) and atomically add into
// agg[dst * rowstride + r*D .. +D).
__global__ void scatter_kernel(const float* __restrict__ feat, int D,
                               const int* __restrict__ ei,
                               float* __restrict__ agg, int rowstride) {
  int gw   = (blockIdx.x * blockDim.x + threadIdx.x) >> 5;
  int lane = threadIdx.x & 31;
  if (gw >= RR * EE) return;
  int r = gw / EE;
  int e = gw - r * EE;
  const int* base = ei + (size_t)r * 2 * EE;
  int src = base[e];
  int dst = base[EE + e];
  const float4* srow = (const float4*)(feat + (size_t)src * D);
  float* drow = agg + (size_t)dst * rowstride + (size_t)r * D;
  int nv = D >> 2;
  for (int i = lane; i < nv; i += 32) {
    float4 v = srow[i];
    atomicAdd(drow + i * 4 + 0, v.x);
    atomicAdd(drow + i * 4 + 1, v.y);
    atomicAdd(drow + i * 4 + 2, v.z);
    atomicAdd(drow + i * 4 + 3, v.w);
  }
}

// ---------------- fused GEMM (+bias, +1/R, optional l2norm+relu) ----------------
// Block = 256 threads = 8 waves, computes 16 nodes x 256 outputs.
// A row = [agg row (Kagg floats) | X row (Kx floats)], K = Kagg + Kx.
// Each wave owns 2 adjacent 16x16 output tiles; K accumulated with
// V_WMMA_F32_16X16X4_F32, A tile staged in LDS per 320-wide chunk.
__global__ __launch_bounds__(256) void gemm_tile_kernel(
    const float* __restrict__ Aagg, int Kagg,
    const float* __restrict__ X, int Kx,
    const float* __restrict__ Bswz,
    const float* __restrict__ bias,
    float* __restrict__ Out, int doNorm)
{
  __shared__ float atile[16 * LDA];
  __shared__ float outb[16 * DOUT_];
  __shared__ float red[16][17];
  __shared__ float scalef[16];

  const int K = Kagg + Kx;
  const int ksteps_total = K >> 2;
  const int t = threadIdx.x;
  const int wave = t >> 5;
  const int lane = t & 31;
  const int mbase = blockIdx.x * 16;
  const int row = lane & 15;
  const int khalf = (lane >> 4) << 1;

  v8f c0 = {};
  v8f c1 = {};
  const int nt0 = wave * 2;
  const int nt1 = wave * 2 + 1;
  const v2f* B0 = (const v2f*)Bswz + (size_t)nt0 * ksteps_total * 32 + lane;
  const v2f* B1 = (const v2f*)Bswz + (size_t)nt1 * ksteps_total * 32 + lane;

  for (int kc0 = 0; kc0 < K; kc0 += KCH) {
    // cooperative, coalesced stage of the 16 x KCH A-tile
    for (int idx = t; idx < 16 * KCH; idx += 256) {
      int m = idx / KCH;
      int k = idx - m * KCH;
      int gk = kc0 + k;
      int node = mbase + m;
      float v = 0.f;
      if (node < NN)
        v = (gk < Kagg) ? Aagg[(size_t)node * Kagg + gk]
                        : X[(size_t)node * Kx + (gk - Kagg)];
      atile[m * LDA + k] = v;
    }
    __syncthreads();

    const int kkbase = kc0 >> 2;
    const float* arow = atile + row * LDA + khalf;
    for (int kk = 0; kk < (KCH >> 2); ++kk) {
      v2f a = *(const v2f*)(arow + kk * 4);
      v2f b0 = B0[(size_t)(kkbase + kk) * 32];
      v2f b1 = B1[(size_t)(kkbase + kk) * 32];
      c0 = __builtin_amdgcn_wmma_f32_16x16x4_f32(false, a, false, b0,
                                                 (short)0, c0, false, false);
      c1 = __builtin_amdgcn_wmma_f32_16x16x4_f32(false, a, false, b1,
                                                 (short)0, c1, false, false);
    }
    __syncthreads();
  }

  // C layout: lanes 0-15 hold N=lane, M=i (VGPR i); lanes 16-31 hold N=lane-16, M=i+8
  int m0 = (lane >> 4) << 3;
  int col0 = wave * 32 + (lane & 15);
  float bv0 = bias[col0];
  float bv1 = bias[col0 + 16];
  for (int i = 0; i < 8; ++i) {
    outb[(m0 + i) * DOUT_ + col0]      = (c0[i] + bv0) * 0.25f;
    outb[(m0 + i) * DOUT_ + col0 + 16] = (c1[i] + bv1) * 0.25f;
  }
  __syncthreads();

  int mrow = t >> 4;
  int j = t & 15;
  int node = mbase + mrow;
  if (doNorm) {
    float s = 0.f;
    for (int q = 0; q < 16; ++q) {
      float v = outb[mrow * DOUT_ + j * 16 + q];
      s += v * v;
    }
    red[mrow][j] = s;
    __syncthreads();
    if (j == 0) {
      float tot = 0.f;
      for (int q = 0; q < 16; ++q) tot += red[mrow][q];
      scalef[mrow] = 1.0f / fmaxf(sqrtf(tot), 1e-12f);
    }
    __syncthreads();
    if (node < NN) {
      float sc = scalef[mrow];
      for (int q = 0; q < 16; ++q) {
        float v = outb[mrow * DOUT_ + j * 16 + q] * sc;
        Out[(size_t)node * DOUT_ + j * 16 + q] = fmaxf(v, 0.f);
      }
    }
  } else {
    if (node < NN) {
      for (int q = 0; q < 16; ++q)
        Out[(size_t)node * DOUT_ + j * 16 + q] = outb[mrow * DOUT_ + j * 16 + q];
    }
  }
}

// ---------------- launch ----------------

extern "C" void kernel_launch(void* const* d_in, const int* in_sizes, int n_in,
                              void* d_out, int out_size, void* d_ws, size_t ws_size,
                              hipStream_t stream) {
  (void)in_sizes; (void)n_in; (void)out_size; (void)ws_size;
  const float* x   = (const float*)d_in[0];
  const int*   ei  = (const int*)  d_in[1];
  const float* W1n = (const float*)d_in[2];
  const float* b1  = (const float*)d_in[3];
  const float* W1r = (const float*)d_in[4];
  const float* W2n = (const float*)d_in[5];
  const float* b2  = (const float*)d_in[6];
  const float* W2r = (const float*)d_in[7];
  float* out = (float*)d_out;
  float* ws  = (float*)d_ws;

  const size_t K1 = (size_t)RR * DIN_ + DIN_;    // 640
  const size_t K2 = (size_t)RR * DOUT_ + DOUT_;  // 1280

  float* Bswz1 = ws;                              // K1*256 floats
  float* Bswz2 = Bswz1 + K1 * DOUT_;              // K2*256 floats
  float* bias1 = Bswz2 + K2 * DOUT_;              // 256
  float* bias2 = bias1 + DOUT_;                   // 256
  float* h     = bias2 + DOUT_;                   // N*256
  float* agg   = h + (size_t)NN * DOUT_;          // N*R*256 (layer1 uses N*R*128)

  // weight prep (swizzled B fragments + bias sums)
  int tot1 = (int)(K1 * DOUT_);   // 163840
  int tot2 = (int)(K2 * DOUT_);   // 327680
  build_bswz_kernel<<<(tot1 + 255) / 256, 256, 0, stream>>>(W1n, W1r, DIN_, Bswz1);
  build_bswz_kernel<<<(tot2 + 255) / 256, 256, 0, stream>>>(W2n, W2r, DOUT_, Bswz2);
  bias_sum_kernel<<<1, 256, 0, stream>>>(b1, bias1);
  bias_sum_kernel<<<1, 256, 0, stream>>>(b2, bias2);

  const int sblocks = (RR * EE + 7) / 8;   // 8 waves (edges) per 256-thread block
  const int gblocks = NN / 16;             // 6250, N divisible by 16

  // ---- layer 1 ----
  zero_kernel<<<4096, 256, 0, stream>>>((float4*)agg, (long)NN * (RR * DIN_) / 4);
  scatter_kernel<<<sblocks, 256, 0, stream>>>(x, DIN_, ei, agg, RR * DIN_);
  gemm_tile_kernel<<<gblocks, 256, 0, stream>>>(agg, RR * DIN_, x, DIN_,
                                                Bswz1, bias1, h, 1);

  // ---- layer 2 ----
  zero_kernel<<<8192, 256, 0, stream>>>((float4*)agg, (long)NN * (RR * DOUT_) / 4);
  scatter_kernel<<<sblocks, 256, 0, stream>>>(h, DOUT_, ei, agg, RR * DOUT_);
  gemm_tile_kernel<<<gblocks, 256, 0, stream>>>(agg, RR * DOUT_, h, DOUT_,
                                                Bswz2, bias2, out, 0);
}